// LiftedStructureLoss_87849261072495
// MI455X (gfx1250) — compile-verified
//
#include <hip/hip_runtime.h>
#include <math.h>

typedef __attribute__((ext_vector_type(2))) float v2f;
typedef __attribute__((ext_vector_type(8))) float v8f;
typedef int v4i_gcc __attribute__((vector_size(16)));   // matches builtin param

#define MARGIN 1.0f
#define DIM 128
#define KSTEPS (DIM / 4)          // 32 wmma f32 16x16x4 per tile
#define TILE 16
#define WAVES_PER_BLOCK 8
#define BLOCK (WAVES_PER_BLOCK * 32)
#define ROWS_PER_BLOCK (WAVES_PER_BLOCK * TILE)   // 128
#define SLICES 8                   // column-slice blocks per 128-row group
#define BSTRIDE 132                // padded LDS row stride (floats): conflict-free
#define CHUNKS (TILE * DIM * 4 / 16)               // 512 x 16B chunks per tile

// ---------------------------------------------------------------------------
// Async staging of one 16x128 fp32 B tile into LDS (tracked by ASYNCcnt).
// ---------------------------------------------------------------------------
__device__ __forceinline__ void issue_tile_copy(const float* __restrict__ E,
                                                int colTile, float* dst) {
#pragma unroll
    for (int c = threadIdx.x; c < CHUNKS; c += BLOCK) {
        const int row   = c >> 5;        // 32 x 16B chunks per 512B row
        const int off16 = c & 31;
        const float* g = E + (size_t)(colTile + row) * DIM + off16 * 4;
        float*       l = dst + row * BSTRIDE + off16 * 4;
#if __has_builtin(__builtin_amdgcn_global_load_async_to_lds_b128)
        __builtin_amdgcn_global_load_async_to_lds_b128(
            (__attribute__((address_space(1))) v4i_gcc*)g,
            (__attribute__((address_space(3))) v4i_gcc*)l,
            0, 0);
#else
        *(float4*)l = *(const float4*)g;      // sync fallback: load + ds_store
#endif
    }
}

__device__ __forceinline__ void wait_tile_copy() {
#if __has_builtin(__builtin_amdgcn_global_load_async_to_lds_b128)
#if __has_builtin(__builtin_amdgcn_s_wait_asynccnt)
    __builtin_amdgcn_s_wait_asynccnt(0);
#else
    asm volatile("s_wait_asynccnt 0x0" ::: "memory");
#endif
#endif
}

// ---------------------------------------------------------------------------
// sq[i] = sum_d E[i,d]^2   (one wave per row; lane loads float4)
// ---------------------------------------------------------------------------
__global__ void lsl_rowsq_kernel(const float* __restrict__ E,
                                 float* __restrict__ sq, int N) {
    const int row  = blockIdx.x * WAVES_PER_BLOCK + (threadIdx.x >> 5);
    const int lane = threadIdx.x & 31;
    if (row >= N) return;
    const float4 v = *(const float4*)(E + (size_t)row * DIM + lane * 4);
    float s = v.x * v.x + v.y * v.y + v.z * v.z + v.w * v.w;
#pragma unroll
    for (int off = 16; off > 0; off >>= 1) s += __shfl_xor(s, off, 32);
    if (lane == 0) sq[row] = s;
}

// ---------------------------------------------------------------------------
// Pass A: sum_exp[i] = sum_{j: label!=, j!=i} exp(MARGIN - dist_ij)
// Block owns 128 rows; 8 waves share an LDS-staged B tile (async double
// buffer, one barrier per tile). Gram tiles via fp32 WMMA.
// ---------------------------------------------------------------------------
__global__ void lsl_sumexp_kernel(const float* __restrict__ E,
                                  const int* __restrict__ labels,
                                  const float* __restrict__ sq,
                                  float* __restrict__ sum_exp, int N) {
    const int nTiles  = N / TILE;
    const int numCT   = nTiles / SLICES;
    const int rowBase = (blockIdx.x / SLICES) * ROWS_PER_BLOCK;
    const int slice   = blockIdx.x % SLICES;
    const int wave = threadIdx.x >> 5;
    const int lane = threadIdx.x & 31;
    const int half = lane >> 4;     // 0: lanes 0-15, 1: lanes 16-31
    const int l16  = lane & 15;
    const int kb   = half * 2;      // A/B VGPR layout: upper half carries K+2
    const int rowTile = rowBase + wave * TILE;

    __shared__ float tileB[2][TILE * BSTRIDE];

    // A operand strip for this wave's 16 rows, register-resident (reused 64x).
    const int arow = rowTile + l16;   // A: lane = M
    v2f a_reg[KSTEPS];
#pragma unroll
    for (int kk = 0; kk < KSTEPS; ++kk) {
        const float2 t = *(const float2*)(E + (size_t)arow * DIM + kk * 4 + kb);
        v2f av; av.x = t.x; av.y = t.y;
        a_reg[kk] = av;
    }

    // Per-lane row metadata (C/D layout: VGPR v -> row = v + 8*half).
    int   rowlab[8];
    float rowsq2[8];
#pragma unroll
    for (int v = 0; v < 8; ++v) {
        const int r = rowTile + v + 8 * half;
        rowlab[v] = labels[r];
        rowsq2[v] = sq[r];
    }

    float acc[8] = {0.f, 0.f, 0.f, 0.f, 0.f, 0.f, 0.f, 0.f};

    issue_tile_copy(E, slice * TILE, tileB[0]);
    for (int i = 0; i < numCT; ++i) {
        const int colTile = (slice + i * SLICES) * TILE;
        wait_tile_copy();
        __syncthreads();                       // tile i ready; tile i-1 drained
        if (i + 1 < numCT)
            issue_tile_copy(E, (slice + (i + 1) * SLICES) * TILE,
                            tileB[(i + 1) & 1]);

        const float* bbase = &tileB[i & 1][l16 * BSTRIDE + kb];
        v8f c = {};
#pragma unroll
        for (int kk = 0; kk < KSTEPS; ++kk) {
            const float2 t = *(const float2*)(bbase + kk * 4);
            v2f bv; bv.x = t.x; bv.y = t.y;
            c = __builtin_amdgcn_wmma_f32_16x16x4_f32(
                    false, a_reg[kk], false, bv, (short)0, c, false, false);
        }

        const int   cidx = colTile + l16;      // C/D layout: N = lane&15
        const int   clab = labels[cidx];
        const float csq  = sq[cidx];
#pragma unroll
        for (int v = 0; v < 8; ++v) {
            const int r = rowTile + v + 8 * half;
            const float d2   = fmaxf(rowsq2[v] + csq - 2.0f * c[v], 0.0f);
            const float dist = sqrtf(d2);
            const float e = __expf(MARGIN - dist);          // ~99% are negatives
            const float m = (r != cidx && rowlab[v] != clab) ? 1.0f : 0.0f;
            acc[v] += m * e;
        }
    }

    // Reduce lane partials; one global atomic per row per block (8 slices/row).
#pragma unroll
    for (int v = 0; v < 8; ++v) {
        float e = acc[v];
#pragma unroll
        for (int off = 1; off < 16; off <<= 1) e += __shfl_xor(e, off, 16);
        if (l16 == 0) atomicAdd(&sum_exp[rowTile + v + 8 * half], e);
    }
}

// ---------------------------------------------------------------------------
// Pass B: loss += relu(log(se_i + se_j) + dist)^2 over positive pairs;
// counts n_pos. Same LDS-shared WMMA tiling.
// ---------------------------------------------------------------------------
__global__ void lsl_loss_kernel(const float* __restrict__ E,
                                const int* __restrict__ labels,
                                const float* __restrict__ sq,
                                const float* __restrict__ sum_exp,
                                float* __restrict__ accum,  // [0]=loss,[1]=npos
                                int N) {
    const int nTiles  = N / TILE;
    const int numCT   = nTiles / SLICES;
    const int rowBase = (blockIdx.x / SLICES) * ROWS_PER_BLOCK;
    const int slice   = blockIdx.x % SLICES;
    const int wave = threadIdx.x >> 5;
    const int lane = threadIdx.x & 31;
    const int half = lane >> 4;
    const int l16  = lane & 15;
    const int kb   = half * 2;
    const int rowTile = rowBase + wave * TILE;

    __shared__ float tileB[2][TILE * BSTRIDE];
    __shared__ float sAcc[2];
    if (threadIdx.x < 2) sAcc[threadIdx.x] = 0.0f;

    const int arow = rowTile + l16;
    v2f a_reg[KSTEPS];
#pragma unroll
    for (int kk = 0; kk < KSTEPS; ++kk) {
        const float2 t = *(const float2*)(E + (size_t)arow * DIM + kk * 4 + kb);
        v2f av; av.x = t.x; av.y = t.y;
        a_reg[kk] = av;
    }

    int   rowlab[8];
    float rowsq2[8];
    float rowse[8];
#pragma unroll
    for (int v = 0; v < 8; ++v) {
        const int r = rowTile + v + 8 * half;
        rowlab[v] = labels[r];
        rowsq2[v] = sq[r];
        rowse[v]  = sum_exp[r];
    }

    float lacc = 0.0f;
    float pcnt = 0.0f;

    issue_tile_copy(E, slice * TILE, tileB[0]);
    for (int i = 0; i < numCT; ++i) {
        const int colTile = (slice + i * SLICES) * TILE;
        wait_tile_copy();
        __syncthreads();
        if (i + 1 < numCT)
            issue_tile_copy(E, (slice + (i + 1) * SLICES) * TILE,
                            tileB[(i + 1) & 1]);

        const float* bbase = &tileB[i & 1][l16 * BSTRIDE + kb];
        v8f c = {};
#pragma unroll
        for (int kk = 0; kk < KSTEPS; ++kk) {
            const float2 t = *(const float2*)(bbase + kk * 4);
            v2f bv; bv.x = t.x; bv.y = t.y;
            c = __builtin_amdgcn_wmma_f32_16x16x4_f32(
                    false, a_reg[kk], false, bv, (short)0, c, false, false);
        }

        const int   cidx = colTile + l16;
        const int   clab = labels[cidx];
        const float csq  = sq[cidx];
        const float cse  = sum_exp[cidx];
#pragma unroll
        for (int v = 0; v < 8; ++v) {
            const int r = rowTile + v + 8 * half;
            if (r != cidx && rowlab[v] == clab) {          // positive pair (~1%)
                const float d2   = fmaxf(rowsq2[v] + csq - 2.0f * c[v], 0.0f);
                const float dist = sqrtf(d2);
                float L = logf(rowse[v] + cse) + dist;
                L = fmaxf(L, 0.0f);
                lacc += L * L;
                pcnt += 1.0f;
            }
        }
    }

    // Wave reduce, block combine, one global atomic pair per block.
#pragma unroll
    for (int off = 16; off > 0; off >>= 1) {
        lacc += __shfl_xor(lacc, off, 32);
        pcnt += __shfl_xor(pcnt, off, 32);
    }
    if (lane == 0) {
        atomicAdd(&sAcc[0], lacc);
        atomicAdd(&sAcc[1], pcnt);
    }
    __syncthreads();
    if (threadIdx.x == 0) {
        atomicAdd(&accum[0], sAcc[0]);
        atomicAdd(&accum[1], sAcc[1]);
    }
    (void)wave;
}

__global__ void lsl_init_kernel(float* __restrict__ accum,
                                float* __restrict__ sum_exp, int N) {
    const int i = blockIdx.x * blockDim.x + threadIdx.x;
    if (i < N) sum_exp[i] = 0.0f;
    if (i < 2) accum[i] = 0.0f;
}

__global__ void lsl_finalize_kernel(const float* __restrict__ accum,
                                    float* __restrict__ out) {
    if (threadIdx.x == 0) out[0] = accum[0] / accum[1] * 0.5f;
}

// ---------------------------------------------------------------------------
extern "C" void kernel_launch(void* const* d_in, const int* in_sizes, int n_in,
                              void* d_out, int out_size, void* d_ws, size_t ws_size,
                              hipStream_t stream) {
    const float* E      = (const float*)d_in[0];
    const int*   labels = (const int*)d_in[1];
    const int N = in_sizes[1];          // 8192; D fixed at 128

    float* ws      = (float*)d_ws;
    float* accum   = ws;                // 2 floats: loss sum, n_pos
    float* sq      = ws + 2;            // N floats
    float* sum_exp = ws + 2 + N;        // N floats
    float* out     = (float*)d_out;

    const int nBlocks = (N / ROWS_PER_BLOCK) * SLICES;   // 512

    lsl_init_kernel<<<(N + BLOCK - 1) / BLOCK, BLOCK, 0, stream>>>(accum, sum_exp, N);
    lsl_rowsq_kernel<<<(N + WAVES_PER_BLOCK - 1) / WAVES_PER_BLOCK, BLOCK, 0, stream>>>(E, sq, N);
    lsl_sumexp_kernel<<<nBlocks, BLOCK, 0, stream>>>(E, labels, sq, sum_exp, N);
    lsl_loss_kernel<<<nBlocks, BLOCK, 0, stream>>>(E, labels, sq, sum_exp, accum, N);
    lsl_finalize_kernel<<<1, 32, 0, stream>>>(accum, out);
}